// YoloV3_LOSS_65790309040613
// MI455X (gfx1250) — compile-verified
//
#include <hip/hip_runtime.h>
#include <hip/hip_bf16.h>

// ---------------------------------------------------------------------------
// YOLOv3 loss for MI455X (gfx1250), compile-only optimized.
//
//  * Only channels 0..4 of each 85-channel anchor matter on the hot path
//    (masks kill everything else except ~0.6% of anchors).
//  * Tensor Data Mover (D#: tile_dim0=5, tensor_dim0_stride=85, 4B elems)
//    DMAs the packed 5-float prefixes of 1024 anchors into LDS per tile.
//    Stride-5-dword LDS reads are bank-conflict free on 64 banks.
//  * Double-buffered TDM pipeline: persistent blocks keep two tensor loads
//    in flight; s_wait_tensorcnt(1) retires the oldest (in-order per wave)
//    while the next tile's DMA overlaps compute.
//  * Masked anchors take a rare slow path reading 80 class channels from
//    global memory.
//  * Deterministic reduction: per-block partials + sequential finalize.
// ---------------------------------------------------------------------------

#define TILE 1024   // anchors per TDM tile
#define BLK  256    // threads per block (8 wave32)
#define NBOX 50     // gt boxes per batch
#define NCLS 80

struct GtRec {
    float tx, ty, tw, th;
    int   cell, best, cls, pad;
};

typedef unsigned int u32x4 __attribute__((ext_vector_type(4)));
typedef int          i32x8 __attribute__((ext_vector_type(8)));
typedef int          i32x4 __attribute__((ext_vector_type(4)));

__device__ __forceinline__ float sigmoidf_fast(float x) {
    return 1.0f / (1.0f + __expf(-x));
}

// Issue one TDM strided-gather load: rows [tileStart, tileStart+TILE) x 5 f32.
__device__ __forceinline__ void tdm_issue(const float* hyp, long long tileStart,
                                          long long N, const float* ldsDst) {
    const unsigned long long gaddr =
        (unsigned long long)(const void*)(hyp + tileStart * 85);
    const unsigned int lds_addr =
        (unsigned int)(unsigned long long)(const void*)ldsDst;

    long long rows = N - tileStart;
    unsigned int td1 = (rows > 0x7FFFFFFFll) ? 0x7FFFFFFFu : (unsigned int)rows;

    // D# group 0: count=1 | lds_addr | global_addr[56:0] | type=2
    u32x4 g0;
    g0.x = 1u;
    g0.y = lds_addr;
    g0.z = (unsigned int)(gaddr & 0xFFFFFFFFull);
    g0.w = (unsigned int)((gaddr >> 32) & 0x1FFFFFFull) | (2u << 30);

    // D# group 1: data_size=4B, tensor_dim0=85, tensor_dim1=rows_left,
    //             tile_dim0=5, tile_dim1=TILE, tensor_dim0_stride=85
    i32x8 g1;
    g1[0] = (int)(2u << 16);                                   // data_size=4B
    g1[1] = (int)((85u & 0xFFFFu) << 16);                      // t_dim0 lo16
    g1[2] = (int)((td1 & 0xFFFFu) << 16);                      // t_dim1 lo16
    g1[3] = (int)(((td1 >> 16) & 0xFFFFu) | (5u << 16));       // t_dim1 hi | tile_dim0
    g1[4] = (int)(TILE & 0xFFFF);                              // tile_dim1 | tile_dim2=0
    g1[5] = 85;                                                // dim0_stride lo32
    g1[6] = 0;
    g1[7] = 0;

    i32x4 gz4 = {0, 0, 0, 0};              // groups 2/3 unused (2D tensor)
    i32x8 gz8 = {0, 0, 0, 0, 0, 0, 0, 0};
    __builtin_amdgcn_tensor_load_to_lds(g0, g1, gz4, gz4, gz8, 0);
}

// ---------------------------------------------------------------------------
// Kernel 1: clear the per-anchor target flag map (word granularity).
// ---------------------------------------------------------------------------
__global__ void yolo_clear(unsigned int* __restrict__ words, long long nwords) {
    long long i = (long long)blockIdx.x * blockDim.x + threadIdx.x;
    long long stride = (long long)gridDim.x * blockDim.x;
    for (; i < nwords; i += stride) words[i] = 0u;
}

// ---------------------------------------------------------------------------
// Kernel 2: build targets. One block per batch; thread 0 runs the 50 boxes
// serially so duplicate (cell,best) hits get last-write-wins order, matching
// one valid ordering of jnp .at[].set scatter.
// ---------------------------------------------------------------------------
__global__ void yolo_build(const float* __restrict__ bbox_gt,
                           const int*   __restrict__ cls_gt,
                           const float* __restrict__ anchors,
                           const int*   __restrict__ scale_p,
                           GtRec*        __restrict__ recs,
                           unsigned char* __restrict__ tmap,
                           int B, int hwA) {
    if (threadIdx.x != 0) return;
    const int b  = blockIdx.x;
    const int sc = scale_p[0];
    const int S  = (sc == 0) ? 13 : ((sc == 1) ? 26 : 52);
    const float* anc = anchors + sc * 6;
    const float r0 = anc[0] / anc[1];
    const float r1 = anc[2] / anc[3];
    const float r2 = anc[4] / anc[5];

    for (int n = 0; n < NBOX; ++n) {
        const float* g = bbox_gt + ((long long)b * NBOX + n) * 4;
        const float gx = g[0], gy = g[1], gw = g[2], gh = g[3];
        const float rb = gw / gh;
        int   best = 0;
        float bd   = fabsf(r0 - rb);
        float d1   = fabsf(r1 - rb);
        if (d1 < bd) { bd = d1; best = 1; }
        float d2   = fabsf(r2 - rb);
        if (d2 < bd) { bd = d2; best = 2; }

        const float cx = gx * (float)S, cy = gy * (float)S;
        const float fx = floorf(cx),    fy = floorf(cy);
        const int   cell = (int)(fy * (float)S + fx);

        GtRec r;
        r.tx = cx - fx; r.ty = cy - fy; r.tw = gw; r.th = gh;
        r.cell = cell;  r.best = best;
        r.cls  = cls_gt[(long long)b * NBOX + n];
        r.pad  = 0;
        recs[b * NBOX + n] = r;

        const long long idx = (long long)b * hwA + (long long)cell * 3 + best;
        if (idx >= 0 && idx < (long long)B * hwA) tmap[idx] = 1;
    }
}

// ---------------------------------------------------------------------------
// Kernel 3: main reduction. Persistent blocks, double-buffered TDM tiles.
// ---------------------------------------------------------------------------
__global__ void __launch_bounds__(BLK)
yolo_main(const float* __restrict__ hyp,
          const float* __restrict__ anchors,
          const int*   __restrict__ scale_p,
          const GtRec* __restrict__ recs,
          const unsigned char* __restrict__ tmap,
          float* __restrict__ partials,
          long long N, int hwA, int ntiles) {
    __shared__ float tileLds[2][TILE * 5];   // 2 x 20 KB TDM destinations
    __shared__ float red[BLK * 3];

    const int sc = scale_p[0];
    const float* anc = anchors + sc * 6;

    float accB = 0.0f, accI = 0.0f, accC = 0.0f;

    int t = blockIdx.x;
    // Prime the pipeline: first tile into buffer 0.
    if (t < ntiles && threadIdx.x == 0)
        tdm_issue(hyp, (long long)t * TILE, N, &tileLds[0][0]);

    int cur = 0;
    for (; t < ntiles; t += gridDim.x) {
        const long long tileStart = (long long)t * TILE;
        const int tn = t + gridDim.x;

        // Warm the flag map for this tile while the DMA completes.
        {
            long long p = tileStart + (long long)threadIdx.x * (TILE / BLK);
            if (p < N) __builtin_prefetch(tmap + p, 0, 0);  // global_prefetch_b8
        }

        if (threadIdx.x == 0) {
            if (tn < ntiles) {
                // Overlap next tile's DMA with this tile's compute; the
                // end-of-iteration barrier below guarantees nobody is still
                // reading the buffer being overwritten.
                tdm_issue(hyp, (long long)tn * TILE, N, &tileLds[cur ^ 1][0]);
                __builtin_amdgcn_s_wait_tensorcnt(1);  // oldest (cur) done
            } else {
                __builtin_amdgcn_s_wait_tensorcnt(0);  // drain
            }
        }
        __syncthreads();   // tile[cur] visible to all waves

        const float* buf = &tileLds[cur][0];
#pragma unroll
        for (int k = 0; k < TILE / BLK; ++k) {
            const int local = threadIdx.x + k * BLK;
            const long long i = tileStart + local;
            if (i >= N) break;

            const float* e = &buf[local * 5];
            const float h0 = e[0], h1 = e[1], h2 = e[2], h3 = e[3], h4 = e[4];

            const int a  = (int)(i % 3);
            const float sx = sigmoidf_fast(h0);
            const float sy = sigmoidf_fast(h1);
            const float w  = __expf(h2) * anc[2 * a];
            const float h  = __expf(h3) * anc[2 * a + 1];
            const float s4 = sigmoidf_fast(h4);

            if (!tmap[i]) {
                // iou_mask = 0.5, iou_built = 0 against the zero box.
                accI = fmaf(0.25f * s4, s4, accI);
            } else {
                const int b    = (int)(i / hwA);
                const int cell = (int)((i % hwA) / 3);

                // Gather matching gt boxes (last-write-wins bbox, class union).
                float tx = 0.f, ty = 0.f, tw = 0.f, th = 0.f;
                unsigned long long clo = 0ull;
                unsigned int       chi = 0u;
                const GtRec* rb = recs + b * NBOX;
                for (int n = 0; n < NBOX; ++n) {
                    const GtRec r = rb[n];
                    if (r.cell == cell && r.best == a) {
                        tx = r.tx; ty = r.ty; tw = r.tw; th = r.th;
                        if (r.cls < 64) clo |= (1ull << r.cls);
                        else            chi |= (1u << (r.cls - 64));
                    }
                }

                // bbox MSE (mask = 1)
                accB += (sx - tx) * (sx - tx) + (sy - ty) * (sy - ty)
                      + (w  - tw) * (w  - tw) + (h  - th) * (h  - th);

                // IoU(pred, built), mask = 1
                const float b1x2 = sx + w,  b1y2 = sy + h;
                const float b2x2 = tx + tw, b2y2 = ty + th;
                const float a1 = w * h, a2 = tw * th;
                const float iw = fmaxf(0.f, fminf(b1x2, b2x2) - fmaxf(sx, tx));
                const float ih = fmaxf(0.f, fminf(b1y2, b2y2) - fmaxf(sy, ty));
                const float inter = iw * ih;
                const float iou = inter / (a1 + a2 - inter);
                accI += (s4 - iou) * (s4 - iou);

                // class MSE vs one-hot union (rare: read 80 ch from global)
                const float* cp = hyp + i * 85 + 5;
                for (int c = 0; c < NCLS; ++c) {
                    const float p = sigmoidf_fast(cp[c]);
                    const float tt = (c < 64) ? (float)((clo >> c) & 1ull)
                                              : (float)((chi >> (c - 64)) & 1u);
                    accC += (p - tt) * (p - tt);
                }
            }
        }
        __syncthreads();   // all waves done with tile[cur] before its reuse
        cur ^= 1;
    }

    // ---- deterministic block reduction ----
    red[threadIdx.x]            = accB;
    red[BLK + threadIdx.x]      = accI;
    red[2 * BLK + threadIdx.x]  = accC;
    __syncthreads();
    for (int s = BLK / 2; s > 0; s >>= 1) {
        if ((int)threadIdx.x < s) {
            red[threadIdx.x]           += red[threadIdx.x + s];
            red[BLK + threadIdx.x]     += red[BLK + threadIdx.x + s];
            red[2 * BLK + threadIdx.x] += red[2 * BLK + threadIdx.x + s];
        }
        __syncthreads();
    }
    if (threadIdx.x == 0) {
        partials[blockIdx.x * 3 + 0] = red[0];
        partials[blockIdx.x * 3 + 1] = red[BLK];
        partials[blockIdx.x * 3 + 2] = red[2 * BLK];
    }
}

// ---------------------------------------------------------------------------
// Kernel 4: deterministic finalize (sequential sum over block partials).
// ---------------------------------------------------------------------------
__global__ void yolo_finalize(const float* __restrict__ partials, int nblocks,
                              long long N, float* __restrict__ out) {
    if (threadIdx.x != 0 || blockIdx.x != 0) return;
    double sb = 0.0, si = 0.0, scs = 0.0;
    for (int i = 0; i < nblocks; ++i) {
        sb  += (double)partials[3 * i + 0];
        si  += (double)partials[3 * i + 1];
        scs += (double)partials[3 * i + 2];
    }
    const double n = (double)N;
    out[0] = (float)(5.0 * (sb / (n * 4.0)) + (si / n) + 0.5 * (scs / (n * 80.0)));
}

// ---------------------------------------------------------------------------
extern "C" void kernel_launch(void* const* d_in, const int* in_sizes, int n_in,
                              void* d_out, int out_size, void* d_ws, size_t ws_size,
                              hipStream_t stream) {
    const float* hyp     = (const float*)d_in[0];
    const float* bbox_gt = (const float*)d_in[1];
    const int*   cls_gt  = (const int*)d_in[2];
    const float* anchors = (const float*)d_in[3];
    const int*   scale_p = (const int*)d_in[4];

    const int  B    = in_sizes[1] / (NBOX * 4);          // 64
    const long long total = (long long)in_sizes[0];      // B*hw*A*85
    const int  perB = (int)(total / B);
    const int  hwA  = perB / 85;                         // hw*A = 8112
    const long long N = (long long)B * hwA;              // total anchors

    // workspace layout
    char* ws = (char*)d_ws;
    GtRec* recs = (GtRec*)(ws + 64);
    size_t off = 64 + (((size_t)B * NBOX * sizeof(GtRec) + 63) & ~(size_t)63);
    unsigned char* tmap = (unsigned char*)(ws + off);
    size_t tmap_bytes = (size_t)N;
    off += (tmap_bytes + 63) & ~(size_t)63;
    float* partials = (float*)(ws + off);

    const int ntiles  = (int)((N + TILE - 1) / TILE);    // 507 for default shape
    const int nblocks = (ntiles < 256) ? ntiles : 256;   // persistent blocks
    const long long nwords = (long long)((tmap_bytes + 3) / 4);

    yolo_clear<<<128, 256, 0, stream>>>((unsigned int*)tmap, nwords);
    yolo_build<<<B, 32, 0, stream>>>(bbox_gt, cls_gt, anchors, scale_p,
                                     recs, tmap, B, hwA);
    yolo_main<<<nblocks, BLK, 0, stream>>>(hyp, anchors, scale_p, recs, tmap,
                                           partials, N, hwA, ntiles);
    yolo_finalize<<<1, 32, 0, stream>>>(partials, nblocks, N, (float*)d_out);
}